// Decoder_88141318848887
// MI455X (gfx1250) — compile-verified
//
#include <hip/hip_runtime.h>
#include <cstdint>
#include <cstddef>

#define HW_   2048
#define EHW_  256
#define NS_   1024
#define TW_   32
#define TH_   8

typedef int v4i_ __attribute__((__vector_size__(4 * sizeof(int))));
typedef __attribute__((address_space(1))) int*  as1_i32;
typedef __attribute__((address_space(3))) int*  as3_i32;
typedef __attribute__((address_space(1))) v4i_* as1_v4i;
typedef __attribute__((address_space(3))) v4i_* as3_v4i;

typedef unsigned u32x4_ __attribute__((__vector_size__(16)));
typedef int      i32x4_ __attribute__((__vector_size__(16)));
typedef int      i32x8_ __attribute__((__vector_size__(32)));

// ---- CDNA5 async global -> LDS helpers (ASYNCcnt path) -------------------

__device__ __forceinline__ void g2l_b32(const void* g, void* l) {
#if __has_builtin(__builtin_amdgcn_global_load_async_to_lds_b32)
  __builtin_amdgcn_global_load_async_to_lds_b32(
      (as1_i32)(uintptr_t)g,
      (as3_i32)(unsigned)(uintptr_t)l, 0, 0);
#else
  asm volatile("global_load_async_to_lds_b32 %0, %1, off"
               :: "v"((unsigned)(uintptr_t)l), "v"(g)
               : "memory");
#endif
}

__device__ __forceinline__ void g2l_b128(const void* g, void* l) {
#if __has_builtin(__builtin_amdgcn_global_load_async_to_lds_b128)
  __builtin_amdgcn_global_load_async_to_lds_b128(
      (as1_v4i)(uintptr_t)g,
      (as3_v4i)(unsigned)(uintptr_t)l, 0, 0);
#else
  asm volatile("global_load_async_to_lds_b128 %0, %1, off"
               :: "v"((unsigned)(uintptr_t)l), "v"(g)
               : "memory");
#endif
}

__device__ __forceinline__ void wait_async0() {
#if __has_builtin(__builtin_amdgcn_s_wait_asynccnt)
  __builtin_amdgcn_s_wait_asynccnt(0);
#else
  asm volatile("s_wait_asynccnt 0" ::: "memory");
#endif
  asm volatile("" ::: "memory");
}

__device__ __forceinline__ void wait_async4() {
#if __has_builtin(__builtin_amdgcn_s_wait_asynccnt)
  __builtin_amdgcn_s_wait_asynccnt(4);
#else
  asm volatile("s_wait_asynccnt 4" ::: "memory");
#endif
  asm volatile("" ::: "memory");
}

__device__ __forceinline__ void wait_tensor0() {
#if __has_builtin(__builtin_amdgcn_s_wait_tensorcnt)
  __builtin_amdgcn_s_wait_tensorcnt(0);
#else
  asm volatile("s_wait_tensorcnt 0x0" ::: "memory");
#endif
  asm volatile("" ::: "memory");
}

// ---- Kernel 1: decode + argmax + bitonic sort by depth -------------------

__global__ __launch_bounds__(1024)
void prep_kernel(const float* __restrict__ data, int4* __restrict__ meta) {
  __shared__ float key[NS_];
  __shared__ int   perm[NS_];
  __shared__ int4  m0[NS_];
  const int i = threadIdx.x;
  const float* row = data + (size_t)i * 69;

  int x = (int)rintf(row[0] * (float)HW_);   // round-half-even, matches jnp.round
  int y = (int)rintf(row[1] * (float)HW_);
  int h = (int)rintf(row[2] * (float)HW_);
  int w = (int)rintf(row[3] * (float)HW_);
  float d = row[4];

  float best = row[5];
  int bi = 0;
  #pragma unroll 8
  for (int j = 1; j < 64; ++j) {
    float v = row[5 + j];
    if (v > best) { best = v; bi = j; }   // first max wins, like jnp.argmax
  }

  int x1 = x - (h >> 1);                   // h >= 0 for these inputs
  int y1 = y - (w >> 1);
  x1 = x1 < 0 ? 0 : (x1 > HW_ - EHW_ ? HW_ - EHW_ : x1);  // dynamic_slice clamp
  y1 = y1 < 0 ? 0 : (y1 > HW_ - EHW_ ? HW_ - EHW_ : y1);

  key[i] = d; perm[i] = i;
  m0[i] = make_int4(x1, y1, bi, 0);
  __syncthreads();

  // Bitonic sort, ascending by depth (matches argsort(d))
  for (int k = 2; k <= NS_; k <<= 1) {
    for (int j = k >> 1; j > 0; j >>= 1) {
      int ixj = i ^ j;
      if (ixj > i) {
        float a = key[i], b = key[ixj];
        bool up = ((i & k) == 0);
        if ((a > b) == up) {
          key[i] = b; key[ixj] = a;
          int t = perm[i]; perm[i] = perm[ixj]; perm[ixj] = t;
        }
      }
      __syncthreads();
    }
  }
  meta[i] = m0[perm[i]];
}

// ---- Kernel 2: per-tile depth-ordered compositing with async LDS staging -

__global__ __launch_bounds__(256)
void raster_kernel(const float* __restrict__ images,
                   const int4* __restrict__ meta,
                   float* __restrict__ out) {
  __shared__ int4  smeta[NS_];             // 16 KB sorted sprite metadata
  __shared__ float sbuf[2][4][256];        // 8 KB double-buffered texels

  const int tid = threadIdx.x;
  const int tb0 = blockIdx.x * TW_;        // column (W) origin of tile
  const int ta0 = blockIdx.y * TH_;        // row (H) origin of tile

#if __has_builtin(__builtin_amdgcn_tensor_load_to_lds)
  // TDM block copy: 4096 DWORDs (16 KB) of metadata -> LDS, one descriptor,
  // issued by wave 0 only; tracked with TENSORcnt.
  if (tid < 32) {
    unsigned lds_base = (unsigned)(uintptr_t)&smeta[0];
    unsigned long long ga = (unsigned long long)(uintptr_t)meta;
    u32x4_ g0;
    g0[0] = 1u;                                            // count=1, user D#
    g0[1] = lds_base;                                      // lds_addr
    g0[2] = (unsigned)ga;                                  // global_addr[31:0]
    g0[3] = (unsigned)((ga >> 32) & 0x01FFFFFFu)           // global_addr[56:32]
          | 0x80000000u;                                   // type=2 ("image")
    i32x8_ g1;
    g1[0] = (int)(2u << 16);                               // data_size=4B
    g1[1] = (int)(4096u << 16);                            // tensor_dim0 lo16
    g1[2] = (int)(1u << 16);                               // tdim0 hi / tensor_dim1=1
    g1[3] = (int)(4096u << 16);                            // tdim1 hi / tile_dim0=4096
    g1[4] = 1;                                             // tile_dim1=1, tile_dim2=0
    g1[5] = 4096;                                          // tensor_dim0_stride lo32
    g1[6] = 0;
    g1[7] = 0;
    i32x4_ z4 = { 0, 0, 0, 0 };
    i32x8_ z8 = { 0, 0, 0, 0, 0, 0, 0, 0 };
    __builtin_amdgcn_tensor_load_to_lds(g0, g1, z4, z4, z8, 0);
    wait_tensor0();
  }
  __syncthreads();
#else
  for (int e = tid; e < NS_; e += 256)
    g2l_b128(&meta[e], &smeta[e]);
  wait_async0();
  __syncthreads();
#endif

  const int a = ta0 + (tid >> 5);          // this thread's canvas row
  const int b = tb0 + (tid & 31);          // this thread's canvas col

  float R = 1.f, G = 1.f, Bl = 1.f, A = 1.f;

  auto composite = [&](int ob, int qx1, int qy1) {
    if (a >= qx1 && a < qx1 + EHW_ && b >= qy1 && b < qy1 + EHW_) {
      float s0 = sbuf[ob][0][tid];
      float s1 = sbuf[ob][1][tid];
      float s2 = sbuf[ob][2][tid];
      float s3 = sbuf[ob][3][tid];
      float an = s3, om = 1.f - an, ao = A;
      float a0 = an + ao * om;
      float inv = 1.f / a0;
      R  = (s0 * an + R  * ao * om) * inv;
      G  = (s1 * an + G  * ao * om) * inv;
      Bl = (s2 * an + Bl * ao * om) * inv;
      A  = a0;
    }
  };

  bool have = false;
  int px1 = 0, py1 = 0, cur = 0;

  for (int r = 0; r < NS_; ++r) {
    int4 m = smeta[r];
    // Uniform tile/sprite bbox test (same for all threads in block)
    if (m.x >= ta0 + TH_ || m.x + EHW_ <= ta0 ||
        m.y >= tb0 + TW_ || m.y + EHW_ <= tb0)
      continue;

    // Stage this sprite's 4 channel texels for this pixel into LDS (async).
    int u = a - m.x; u = u < 0 ? 0 : (u > EHW_ - 1 ? EHW_ - 1 : u);
    int v = b - m.y; v = v < 0 ? 0 : (v > EHW_ - 1 ? EHW_ - 1 : v);
    const float* sp = images + (((size_t)(unsigned)m.z) << 18) + (u << 8) + v;
    g2l_b32(sp,             &sbuf[cur][0][tid]);
    g2l_b32(sp + (1 << 16), &sbuf[cur][1][tid]);
    g2l_b32(sp + (2 << 16), &sbuf[cur][2][tid]);
    g2l_b32(sp + (3 << 16), &sbuf[cur][3][tid]);

    if (have) {
      wait_async4();                 // previous sprite's 4 loads complete
      composite(cur ^ 1, px1, py1);
    }
    have = true; px1 = m.x; py1 = m.y; cur ^= 1;
  }

  if (have) {
    wait_async0();
    composite(cur ^ 1, px1, py1);
  }

  size_t o = ((size_t)a << 11) + (size_t)b;
  out[o]                      = R;
  out[o + ((size_t)1 << 22)]  = G;
  out[o + ((size_t)2 << 22)]  = Bl;
  out[o + ((size_t)3 << 22)]  = A;
}

// ---- Launch ---------------------------------------------------------------

extern "C" void kernel_launch(void* const* d_in, const int* in_sizes, int n_in,
                              void* d_out, int out_size, void* d_ws, size_t ws_size,
                              hipStream_t stream) {
  const float* data   = (const float*)d_in[0];   // (1024, 69)
  const float* images = (const float*)d_in[1];   // (64, 4, 256, 256)
  int4* meta = (int4*)d_ws;                      // 16 KB sorted metadata

  prep_kernel<<<1, 1024, 0, stream>>>(data, meta);

  dim3 grid(HW_ / TW_, HW_ / TH_);               // 64 x 256 tiles of 32x8
  raster_kernel<<<grid, 256, 0, stream>>>(images, meta, (float*)d_out);
}